// GraphCVAEEncoder_1099511628360
// MI455X (gfx1250) — compile-verified
//
#include <hip/hip_runtime.h>
#include <hip/hip_bf16.h>

#define NN   50000
#define EE   800000
#define DIN_ 128
#define DOUT_ 64
#define H1_  128
#define H2_  256
#define H3_  512
#define BN_EPS_ 1e-5f

typedef float v2f __attribute__((ext_vector_type(2)));
typedef float v8f __attribute__((ext_vector_type(8)));

// ---------------------------------------------------------------------------
// C[nrows x M] = A[nrows x K] @ B[K x M] (+ bias).  TRANSB: B is [M x K] row
// major and we use B^T.  Exact fp32 via V_WMMA_F32_16X16X4_F32 (wave32).
// block = 128 threads (4 waves); wave w -> rows [r0+16w, +16), cols [c0, c0+64).
// B is staged per 16-deep K-chunk into LDS (transposed) and shared by all 4
// waves: inner loop = 1 global_load_b64 (A) + 4 ds_load_b64 (B) + 4 wmma.
// ---------------------------------------------------------------------------
template <int TRANSB>
__global__ void wmma_gemm_f32(const float* __restrict__ A,
                              const float* __restrict__ B,
                              const float* __restrict__ bias,
                              float* __restrict__ C,
                              int nrows, int K, int M, int hasBias)
{
    // [col][k] transposed B chunk; row stride 18 floats keeps ds_load_b64
    // 8B-aligned (even float offset) and staggers banks.
    __shared__ float bT[64][18];

    const int wave = threadIdx.x >> 5;
    const int lane = threadIdx.x & 31;
    const int half = lane >> 4;      // 0: lanes 0-15, 1: lanes 16-31
    const int l15  = lane & 15;
    const int r0 = blockIdx.x * 64 + wave * 16;
    const int c0 = blockIdx.y * 64;

    // A fragment: lane holds row (r0+l15), K-pair (k+2*half, k+2*half+1)
    int aRow = r0 + l15;
    if (aRow >= nrows) aRow = nrows - 1;       // clamp; stores are guarded
    const int kOff = half * 2;
    const float* aPtr = A + (size_t)aRow * K + kOff;

    v8f acc0 = {}, acc1 = {}, acc2 = {}, acc3 = {};

    for (int k0 = 0; k0 < K; k0 += 16) {
        __syncthreads();
        // cooperative stage of B[k0:k0+16, c0:c0+64] into bT (transposed)
        #pragma unroll
        for (int i = 0; i < 8; ++i) {
            int idx = threadIdx.x + i * 128;       // 0..1023
            int kk  = idx >> 6;
            int col = idx & 63;
            float v;
            if (TRANSB) v = B[(size_t)(c0 + col) * K + k0 + kk];
            else        v = B[(size_t)(k0 + kk) * M + c0 + col];
            bT[col][kk] = v;
        }
        __syncthreads();

        // speculative prefetch of the next A chunk (global_prefetch_b8)
        __builtin_prefetch(aPtr + k0 + 16, 0, 1);

        #pragma unroll
        for (int kk = 0; kk < 16; kk += 4) {
            v2f a  = *(const v2f*)(aPtr + k0 + kk);
            v2f b0 = *(const v2f*)&bT[l15     ][kk + kOff];
            v2f b1 = *(const v2f*)&bT[l15 + 16][kk + kOff];
            v2f b2 = *(const v2f*)&bT[l15 + 32][kk + kOff];
            v2f b3 = *(const v2f*)&bT[l15 + 48][kk + kOff];
            acc0 = __builtin_amdgcn_wmma_f32_16x16x4_f32(false, a, false, b0, (short)0, acc0, false, false);
            acc1 = __builtin_amdgcn_wmma_f32_16x16x4_f32(false, a, false, b1, (short)0, acc1, false, false);
            acc2 = __builtin_amdgcn_wmma_f32_16x16x4_f32(false, a, false, b2, (short)0, acc2, false, false);
            acc3 = __builtin_amdgcn_wmma_f32_16x16x4_f32(false, a, false, b3, (short)0, acc3, false, false);
        }
    }

    // C/D layout: VGPR r -> row r0 + 8*half + r, lane l15 -> col
    const int rowBase = r0 + half * 8;
    v8f accs[4] = {acc0, acc1, acc2, acc3};
    #pragma unroll
    for (int t = 0; t < 4; ++t) {
        const int col = c0 + t * 16 + l15;
        const float bv = hasBias ? bias[col] : 0.0f;
        #pragma unroll
        for (int r = 0; r < 8; ++r) {
            const int row = rowBase + r;
            if (row < nrows) C[(size_t)row * M + col] = accs[t][r] + bv;
        }
    }
}

// ---------------------------------------------------------------------------
// Degree / dinv
// ---------------------------------------------------------------------------
__global__ void degree_kernel(const int* __restrict__ dst, float* __restrict__ deg, int e)
{
    int i = blockIdx.x * blockDim.x + threadIdx.x;
    if (i < e) atomicAdd(&deg[dst[i]], 1.0f);
}

__global__ void dinv_kernel(float* __restrict__ deg, int n)
{
    int i = blockIdx.x * blockDim.x + threadIdx.x;
    if (i < n) deg[i] = rsqrtf(deg[i] + 1.0f);   // in-place deg -> dinv
}

// ---------------------------------------------------------------------------
// agg[dst] += lin[src] * dinv[src]*dinv[dst]   (thread per edge-channel)
// ---------------------------------------------------------------------------
__global__ void scatter_edges(const int* __restrict__ srcI, const int* __restrict__ dstI,
                              const float* __restrict__ lin, const float* __restrict__ dinv,
                              float* __restrict__ agg, int e, int cbits)
{
    const int C = 1 << cbits;
    long long i = (long long)blockIdx.x * blockDim.x + threadIdx.x;
    if (i >= ((long long)e << cbits)) return;
    int ed = (int)(i >> cbits);
    int c  = (int)(i & (C - 1));
    int s  = srcI[ed];
    int d  = dstI[ed];
    float nrm = dinv[s] * dinv[d];
    atomicAdd(&agg[(size_t)d * C + c], lin[(size_t)s * C + c] * nrm);
}

// ---------------------------------------------------------------------------
// t = agg + dinv^2 * lin + bias  (in place into agg), accumulate BN column
// sums/sumsq via LDS atomics then one global atomic flush per column.
// grid.x covers 64-row strips.
// ---------------------------------------------------------------------------
__global__ void gcn_finish_stats(float* __restrict__ agg, const float* __restrict__ lin,
                                 const float* __restrict__ dinv, const float* __restrict__ bias,
                                 float* __restrict__ colsum, float* __restrict__ colsq,
                                 int nrows, int C)
{
    __shared__ float ssum[512];
    __shared__ float ssq[512];
    for (int c = threadIdx.x; c < C; c += blockDim.x) { ssum[c] = 0.0f; ssq[c] = 0.0f; }
    __syncthreads();

    const int r0 = blockIdx.x * 64;
    int rend = r0 + 64; if (rend > nrows) rend = nrows;
    const int cnt = (rend - r0) * C;
    for (int idx = threadIdx.x; idx < cnt; idx += blockDim.x) {
        int r = r0 + idx / C;
        int c = idx % C;
        size_t off = (size_t)r * C + c;
        float di = dinv[r];
        float t = agg[off] + di * di * lin[off] + bias[c];
        agg[off] = t;
        atomicAdd(&ssum[c], t);
        atomicAdd(&ssq[c], t * t);
    }
    __syncthreads();
    for (int c = threadIdx.x; c < C; c += blockDim.x) {
        atomicAdd(&colsum[c], ssum[c]);
        atomicAdd(&colsq[c],  ssq[c]);
    }
}

// ---------------------------------------------------------------------------
// out = mode ? elu(bn(t) + h0) : bn(t)
// ---------------------------------------------------------------------------
__global__ void bn_finalize(const float* __restrict__ t,
                            const float* __restrict__ colsum, const float* __restrict__ colsq,
                            const float* __restrict__ gamma, const float* __restrict__ beta,
                            const float* __restrict__ h0, float* __restrict__ out,
                            int nrows, int C, int mode)
{
    long long i = (long long)blockIdx.x * blockDim.x + threadIdx.x;
    if (i >= (long long)nrows * C) return;
    int c = (int)(i % C);
    float inv_n = 1.0f / (float)nrows;
    float mean = colsum[c] * inv_n;
    float var  = colsq[c] * inv_n - mean * mean;
    float v = (t[i] - mean) * rsqrtf(var + BN_EPS_) * gamma[c] + beta[c];
    if (mode) {
        v += h0[i];
        v = v > 0.0f ? v : (expf(v) - 1.0f);
    }
    out[i] = v;
}

__global__ void reparam_kernel(const float* __restrict__ mu, const float* __restrict__ lv,
                               const float* __restrict__ eps, float* __restrict__ z, long long n)
{
    long long i = (long long)blockIdx.x * blockDim.x + threadIdx.x;
    if (i < n) z[i] = mu[i] + eps[i] * expf(0.5f * lv[i]);
}

// one wave32 per row
__global__ void l2norm_rows(const float* __restrict__ z, float* __restrict__ g, int nrows, int C)
{
    int wave = threadIdx.x >> 5, lane = threadIdx.x & 31;
    int row = blockIdx.x * 4 + wave;
    if (row >= nrows) return;
    float s = 0.0f;
    for (int c = lane; c < C; c += 32) { float v = z[(size_t)row * C + c]; s += v * v; }
    #pragma unroll
    for (int o = 16; o; o >>= 1) s += __shfl_xor(s, o, 32);
    float inv = 1.0f / fmaxf(sqrtf(s), 1e-12f);
    for (int c = lane; c < C; c += 32) g[(size_t)row * C + c] = z[(size_t)row * C + c] * inv;
}

// ret[n]   = [ z.Wg + b,  g_a.Wg + b ]
// ret_a[n] = [ g_a.Wga + b, z.Wga + b ]
__global__ void disc_kernel(const float* __restrict__ z, const float* __restrict__ g_a,
                            const float* __restrict__ Wg, const float* __restrict__ Wga,
                            const float* __restrict__ db, float* __restrict__ ret,
                            float* __restrict__ ret_a, int nrows)
{
    int wave = threadIdx.x >> 5, lane = threadIdx.x & 31;
    int row = blockIdx.x * 4 + wave;
    if (row >= nrows) return;
    float d1 = 0, d2 = 0, d3 = 0, d4 = 0;
    for (int c = lane; c < DOUT_; c += 32) {
        size_t o = (size_t)row * DOUT_ + c;
        float zv = z[o], ga = g_a[o], wg = Wg[o], wga = Wga[o];
        d1 += zv * wg;  d2 += ga * wg;  d3 += ga * wga;  d4 += zv * wga;
    }
    #pragma unroll
    for (int o = 16; o; o >>= 1) {
        d1 += __shfl_xor(d1, o, 32); d2 += __shfl_xor(d2, o, 32);
        d3 += __shfl_xor(d3, o, 32); d4 += __shfl_xor(d4, o, 32);
    }
    if (lane == 0) {
        float b = db[0];
        ret[2 * row]     = d1 + b;  ret[2 * row + 1]   = d2 + b;
        ret_a[2 * row]   = d3 + b;  ret_a[2 * row + 1] = d4 + b;
    }
}

// ---------------------------------------------------------------------------
extern "C" void kernel_launch(void* const* d_in, const int* in_sizes, int n_in,
                              void* d_out, int out_size, void* d_ws, size_t ws_size,
                              hipStream_t stream)
{
    (void)in_sizes; (void)n_in; (void)out_size; (void)ws_size;
    const float* x    = (const float*)d_in[0];
    const float* x_a  = (const float*)d_in[1];
    const int*   ei   = (const int*)d_in[2];
    const float* eps1 = (const float*)d_in[3];
    const float* eps2 = (const float*)d_in[4];
    const float* w1 = (const float*)d_in[5];   const float* b1 = (const float*)d_in[6];
    const float* w2 = (const float*)d_in[7];   const float* b2 = (const float*)d_in[8];
    const float* w3 = (const float*)d_in[9];   const float* b3 = (const float*)d_in[10];
    const float* wm = (const float*)d_in[11];  const float* bm = (const float*)d_in[12];
    const float* wl = (const float*)d_in[13];  const float* bl = (const float*)d_in[14];
    const float* g1 = (const float*)d_in[15];  const float* be1 = (const float*)d_in[16];
    const float* g2 = (const float*)d_in[17];  const float* be2 = (const float*)d_in[18];
    const float* g3 = (const float*)d_in[19];  const float* be3 = (const float*)d_in[20];
    const float* gm = (const float*)d_in[21];  const float* bem = (const float*)d_in[22];
    const float* gl = (const float*)d_in[23];  const float* bel = (const float*)d_in[24];
    const float* pw1 = (const float*)d_in[25]; const float* pb1 = (const float*)d_in[26];
    const float* pw2 = (const float*)d_in[27]; const float* pb2 = (const float*)d_in[28];
    const float* pw3 = (const float*)d_in[29]; const float* pb3 = (const float*)d_in[30];
    const float* fw = (const float*)d_in[31];  const float* fb = (const float*)d_in[32];
    const float* dw = (const float*)d_in[33];  const float* db = (const float*)d_in[34];

    const int* srcI = ei;
    const int* dstI = ei + EE;

    // ---- workspace layout (floats) ----
    const size_t BIG = (size_t)NN * H3_;     // 25.6M
    const size_t N64 = (size_t)NN * DOUT_;   // 3.2M
    float* ws    = (float*)d_ws;
    float* dinv  = ws;                       // NN
    float* hA    = ws + 65536;               // activation ping
    float* hB    = hA + BIG;                 // activation pong
    float* lin   = hB + BIG;                 // GCN linear output
    float* h0b   = lin + BIG;                // residual projection
    float* agg   = h0b + BIG;                // scatter target / t
    float* colsum = agg + BIG;               // 512
    float* colsq  = colsum + 512;            // 512
    float* mu_a  = colsq + 512;
    float* lv_a  = mu_a + N64;
    float* zbuf  = lv_a + N64;
    float* gbuf  = zbuf + N64;
    float* z_a   = gbuf + N64;
    float* g_a   = z_a + N64;
    float* Wg    = g_a + N64;
    float* Wga   = Wg + N64;

    // ---- d_out layout: mu, logvar, hidden, reconstructed, ret, ret_a ----
    float* out       = (float*)d_out;
    float* out_mu    = out;
    float* out_lv    = out_mu + N64;
    float* out_hid   = out_lv + N64;
    float* out_recon = out_hid + BIG;
    float* out_ret   = out_recon + (size_t)NN * DIN_;
    float* out_reta  = out_ret + 2 * (size_t)NN;

    // ---- degree / dinv ----
    hipMemsetAsync(dinv, 0, NN * sizeof(float), stream);
    degree_kernel<<<(EE + 255) / 256, 256, 0, stream>>>(dstI, dinv, EE);
    dinv_kernel<<<(NN + 255) / 256, 256, 0, stream>>>(dinv, NN);

    auto gemm = [&](const float* A, const float* B, const float* bias, float* C,
                    int K, int M, int transB) {
        dim3 grid((NN + 63) / 64, M / 64);
        if (transB)
            wmma_gemm_f32<1><<<grid, 128, 0, stream>>>(A, B, bias, C, NN, K, M,
                                                       bias != nullptr ? 1 : 0);
        else
            wmma_gemm_f32<0><<<grid, 128, 0, stream>>>(A, B, bias, C, NN, K, M,
                                                       bias != nullptr ? 1 : 0);
    };

    auto gcn_layer = [&](int C, const float* bias, const float* gamma, const float* beta,
                         const float* h0p, float* outp, int mode) {
        int cbits = __builtin_ctz(C);
        hipMemsetAsync(agg, 0, (size_t)NN * C * sizeof(float), stream);
        long long tot = (long long)EE << cbits;
        scatter_edges<<<(unsigned)((tot + 255) / 256), 256, 0, stream>>>(
            srcI, dstI, lin, dinv, agg, EE, cbits);
        hipMemsetAsync(colsum, 0, 1024 * sizeof(float), stream);
        gcn_finish_stats<<<(NN + 63) / 64, 256, 0, stream>>>(agg, lin, dinv, bias,
                                                             colsum, colsq, NN, C);
        long long ne = (long long)NN * C;
        bn_finalize<<<(unsigned)((ne + 255) / 256), 256, 0, stream>>>(
            agg, colsum, colsq, gamma, beta, h0p, outp, NN, C, mode);
    };

    auto encode = [&](const float* xin, float* muo, float* lvo, float* hido) {
        // layer 1
        gemm(xin, w1, nullptr, lin, DIN_, H1_, 0);
        gemm(xin, pw1, pb1, h0b, DIN_, H1_, 0);
        gcn_layer(H1_, b1, g1, be1, h0b, hA, 1);             // hA = h1
        // layer 2 (h0_2 kept in h0b for layer 3)
        gemm(hA, w2, nullptr, lin, H1_, H2_, 0);
        gemm(hA, pw2, pb2, h0b, H1_, H2_, 0);
        gcn_layer(H2_, b2, g2, be2, h0b, hB, 1);             // hB = h2
        // layer 3 — note reference: h0_3 = h0_2 @ pw3 + pb3
        gemm(hB, w3, nullptr, lin, H2_, H3_, 0);
        gemm(h0b, pw3, pb3, hA, H2_, H3_, 0);                // hA = h0_3 (h1 dead)
        gcn_layer(H3_, b3, g3, be3, hA, hido, 1);            // hido = hidden
        // mu / logvar heads (bn only)
        gemm(hido, wm, nullptr, lin, H3_, DOUT_, 0);
        gcn_layer(DOUT_, bm, gm, bem, nullptr, muo, 0);
        gemm(hido, wl, nullptr, lin, H3_, DOUT_, 0);
        gcn_layer(DOUT_, bl, gl, bel, nullptr, lvo, 0);
    };

    // encoder 1 (x)
    encode(x, out_mu, out_lv, out_hid);
    long long n64 = (long long)NN * DOUT_;
    reparam_kernel<<<(unsigned)((n64 + 255) / 256), 256, 0, stream>>>(out_mu, out_lv, eps1, zbuf, n64);
    gemm(zbuf, fw, fb, out_recon, DOUT_, DIN_, 0);           // reconstructed
    l2norm_rows<<<(NN + 3) / 4, 128, 0, stream>>>(zbuf, gbuf, NN, DOUT_);

    // encoder 2 (x_a) — hidden_a scratch in hB
    encode(x_a, mu_a, lv_a, hB);
    reparam_kernel<<<(unsigned)((n64 + 255) / 256), 256, 0, stream>>>(mu_a, lv_a, eps2, z_a, n64);
    l2norm_rows<<<(NN + 3) / 4, 128, 0, stream>>>(z_a, g_a, NN, DOUT_);

    // discriminator: Wg = g @ dw^T, Wga = g_a @ dw^T, then 4 row-dots
    gemm(gbuf, dw, nullptr, Wg, DOUT_, DOUT_, 1);
    gemm(g_a,  dw, nullptr, Wga, DOUT_, DOUT_, 1);
    disc_kernel<<<(NN + 3) / 4, 128, 0, stream>>>(zbuf, g_a, Wg, Wga, db, out_ret, out_reta, NN);
}